// IKDLoss_85435489452263
// MI455X (gfx1250) — compile-verified
//
#include <hip/hip_runtime.h>
#include <hip/hip_bf16.h>

typedef __attribute__((ext_vector_type(2))) float v2f;
typedef __attribute__((ext_vector_type(8))) float v8f;

struct LvlStats {
  double sum_d;
  double sum_d2;
  double sum_masked;
  double sq;
  float  margin;
  float  nmask;
  float  mu;
  float  _pad;
};

// ---------------------------------------------------------------- zero ws
__global__ void k_zero(float* p, size_t n) {
  size_t i = (size_t)blockIdx.x * blockDim.x + threadIdx.x;
  size_t stride = (size_t)gridDim.x * blockDim.x;
  for (; i < n; i += stride) p[i] = 0.f;
}

// --------------------------------------------- pass 1: norms + d + stats
// One thread per pixel; loop over channels (coalesced: consecutive threads
// hit consecutive hw within a channel plane).
__global__ void k_norm_d(const float* __restrict__ fe, const float* __restrict__ fa,
                         float* __restrict__ rne, float* __restrict__ rna,
                         float* __restrict__ dd, LvlStats* st,
                         int C, int hwShift) {
  const int p = blockIdx.x * 256 + threadIdx.x;
  const int HWm1 = (1 << hwShift) - 1;
  const int b = p >> hwShift;
  const int hw = p & HWm1;
  const size_t base = (((size_t)b * C) << hwShift) + hw;
  float se = 0.f, sa = 0.f, sea = 0.f;
  for (int c = 0; c < C; ++c) {
    float e = fe[base + ((size_t)c << hwShift)];
    float a = fa[base + ((size_t)c << hwShift)];
    se = fmaf(e, e, se);
    sa = fmaf(a, a, sa);
    sea = fmaf(e, a, sea);
  }
  const float rne_ = 1.f / fmaxf(sqrtf(se), 1e-12f);
  const float rna_ = 1.f / fmaxf(sqrtf(sa), 1e-12f);
  const float d = se * rne_ * rne_ - 2.f * sea * rne_ * rna_ + sa * rna_ * rna_;
  rne[p] = rne_;
  rna[p] = rna_;
  dd[p] = d;

  __shared__ double s1[256];
  __shared__ double s2[256];
  s1[threadIdx.x] = (double)d;
  s2[threadIdx.x] = (double)d * (double)d;
  __syncthreads();
  for (int off = 128; off > 0; off >>= 1) {
    if ((int)threadIdx.x < off) {
      s1[threadIdx.x] += s1[threadIdx.x + off];
      s2[threadIdx.x] += s2[threadIdx.x + off];
    }
    __syncthreads();
  }
  if (threadIdx.x == 0) {
    atomicAdd(&st->sum_d, s1[0]);
    atomicAdd(&st->sum_d2, s2[0]);
  }
}

// ------------------------------------------------------- margin (1 thread)
__global__ void k_margin(LvlStats* st, int N) {
  double mu = st->sum_d / (double)N;
  double var = (st->sum_d2 - (double)N * mu * mu) / (double)(N - 1);
  double sd = sqrt(fmax(var, 0.0));
  st->margin = (float)(0.99 * (mu + 2.0 * sd));  // (1-ALPHA)*(mu + BETA*std)
  st->mu = (float)mu;
}

// --------------------------- mask: fold mask into stored reciprocal norms
__global__ void k_mask(const float* __restrict__ dd,
                       float* __restrict__ rne, float* __restrict__ rna,
                       LvlStats* st) {
  const int p = blockIdx.x * 256 + threadIdx.x;
  const float m = st->margin;
  const float d = dd[p];
  const bool sel = d >= m;
  if (!sel) { rne[p] = 0.f; rna[p] = 0.f; }

  __shared__ double ssum[256];
  __shared__ float scnt[256];
  ssum[threadIdx.x] = sel ? (double)d : 0.0;
  scnt[threadIdx.x] = sel ? 1.f : 0.f;
  __syncthreads();
  for (int off = 128; off > 0; off >>= 1) {
    if ((int)threadIdx.x < off) {
      ssum[threadIdx.x] += ssum[threadIdx.x + off];
      scnt[threadIdx.x] += scnt[threadIdx.x + off];
    }
    __syncthreads();
  }
  if (threadIdx.x == 0) {
    atomicAdd(&st->sum_masked, ssum[0]);
    atomicAdd(&st->nmask, scnt[0]);
  }
}

// ----------------------------------------------- Gram via f32 WMMA 16x16x4
// Grid: (C/64, C/64, N/1024). 512 threads = 16 waves; wave (wi,wj) owns the
// 16x16 WMMA tile at (i0+16wi, j0+16wj). LDS tiles are channel-major
// [64ch][64px] with a (+4*ch mod 64) column swizzle -> conflict-free
// ds_load_b64 of (K,K+1) pairs matching the 16x16x4 f32 A/B layouts.
#define GTPX 64
__global__ __launch_bounds__(512, 1)
void k_gram(const float* __restrict__ fe, const float* __restrict__ fa,
            const float* __restrict__ sce, const float* __restrict__ sca,
            float* __restrict__ Gme, float* __restrict__ Gxa, float* __restrict__ Gma,
            int C, int hwShift, int pixPerChunk) {
  __shared__ float Tei[64][GTPX];
  __shared__ float Tai[64][GTPX];
  __shared__ float Tej[64][GTPX];
  __shared__ float Taj[64][GTPX];

  const int i0 = blockIdx.x * 64;
  const int j0 = blockIdx.y * 64;
  const int p0 = blockIdx.z * pixPerChunk;
  const int tid = threadIdx.x;
  const int lane = tid & 31;
  const int wave = tid >> 5;
  const int half = lane >> 4;   // K-pair selector: 0 -> K0/K1, 1 -> K2/K3
  const int l16 = lane & 15;
  const int wi = wave >> 2;
  const int wj = wave & 3;
  const int HWm1 = (1 << hwShift) - 1;
  const int tp = tid & 63;      // pixel within staging tile
  const int tc0 = tid >> 6;     // channel base (0..7, step 8)

  v8f accM = {};
  v8f accX = {};
  v8f accA = {};

  for (int pc = 0; pc < pixPerChunk; pc += 64) {
    const int p = p0 + pc + tp;
    const int b = p >> hwShift;
    const int hw = p & HWm1;
    const size_t base = (((size_t)b * C) << hwShift) + hw;
    const float se = sce[p];   // 0 if pixel not selected
    const float sa = sca[p];
    __syncthreads();
#pragma unroll
    for (int ii = 0; ii < 8; ++ii) {
      const int cr = tc0 + 8 * ii;
      const int col = (tp + 4 * cr) & 63;  // swizzled column
      const size_t offi = base + ((size_t)(i0 + cr) << hwShift);
      const size_t offj = base + ((size_t)(j0 + cr) << hwShift);
      Tei[cr][col] = fe[offi] * se;
      Tai[cr][col] = fa[offi] * sa;
      Tej[cr][col] = fe[offj] * se;
      Taj[cr][col] = fa[offj] * sa;
    }
    __syncthreads();
#pragma unroll
    for (int kk = 0; kk < 64; kk += 4) {
      const int kb = kk + 2 * half;
      const int sc = (kb + 4 * l16) & 63;        // swizzled K-pair column
      const int rowA = 16 * wi + l16;
      const int rowB = 16 * wj + l16;
      v2f ae = *(const v2f*)&Tei[rowA][sc];
      v2f aa = *(const v2f*)&Tai[rowA][sc];
      v2f be = *(const v2f*)&Tej[rowB][sc];
      v2f ba = *(const v2f*)&Taj[rowB][sc];
      accM = __builtin_amdgcn_wmma_f32_16x16x4_f32(false, ae, false, be, (short)0, accM, false, false);
      accX = __builtin_amdgcn_wmma_f32_16x16x4_f32(false, ae, false, ba, (short)0, accX, false, false);
      accA = __builtin_amdgcn_wmma_f32_16x16x4_f32(false, aa, false, ba, (short)0, accA, false, false);
    }
  }

  // D layout: VGPR r, lanes 0-15 -> (M=r, N=lane); lanes 16-31 -> (M=r+8).
  const int col = j0 + 16 * wj + l16;
#pragma unroll
  for (int r = 0; r < 8; ++r) {
    const int row = i0 + 16 * wi + r + 8 * half;
    atomicAdd(&Gme[(size_t)row * C + col], accM[r]);
    atomicAdd(&Gxa[(size_t)row * C + col], accX[r]);
    atomicAdd(&Gma[(size_t)row * C + col], accA[r]);
  }
}

// ------------------------------------- ||Me||^2 - 2||Xea||^2 + ||Ma||^2
__global__ void k_gram_reduce(const float* __restrict__ Gme,
                              const float* __restrict__ Gxa,
                              const float* __restrict__ Gma,
                              LvlStats* st, int total) {
  int i = blockIdx.x * 256 + threadIdx.x;
  const int stride = gridDim.x * 256;
  double part = 0.0;
  for (; i < total; i += stride) {
    double me = Gme[i], xa = Gxa[i], ma = Gma[i];
    part += me * me - 2.0 * xa * xa + ma * ma;
  }
  __shared__ double s[256];
  s[threadIdx.x] = part;
  __syncthreads();
  for (int off = 128; off > 0; off >>= 1) {
    if ((int)threadIdx.x < off) s[threadIdx.x] += s[threadIdx.x + off];
    __syncthreads();
  }
  if (threadIdx.x == 0) atomicAdd(&st->sq, s[0]);
}

// ----------------------------------------------------------------- final
__global__ void k_final(const LvlStats* s0, const LvlStats* s1,
                        const LvlStats* s2, float* out) {
  const LvlStats* sts[3] = {s0, s1, s2};
  double loss = 0.0;
  for (int l = 0; l < 3; ++l) {
    const LvlStats* s = sts[l];
    double n = (double)s->nmask;
    double l_ps = (n > 0.0) ? (s->sum_masked / n) : (double)s->mu;
    double l_cs = (n > 0.0) ? (s->sq / (n * n)) : 0.0;
    loss += l_ps + 1.0 * l_cs;   // GAMMA = 1.0
  }
  out[0] = (float)loss;
}

// ================================================================= launch
extern "C" void kernel_launch(void* const* d_in, const int* in_sizes, int n_in,
                              void* d_out, int out_size, void* d_ws, size_t ws_size,
                              hipStream_t stream) {
  (void)in_sizes; (void)n_in; (void)out_size; (void)ws_size;
  static const int Cs[3] = {256, 512, 1024};
  static const int hwShifts[3] = {12, 10, 8};          // HW = 4096, 1024, 256
  static const int Ns[3] = {8 * 4096, 8 * 1024, 8 * 256};  // 32768, 8192, 2048

  char* ws = (char*)d_ws;
  size_t off = 0;
  LvlStats* stats = (LvlStats*)ws;
  off += 256;  // 3 structs (48B each), padded

  float* gram[3][3];
  for (int l = 0; l < 3; ++l)
    for (int m = 0; m < 3; ++m) {
      gram[l][m] = (float*)(ws + off);
      off += (size_t)Cs[l] * Cs[l] * 4;
    }
  const size_t zero_bytes = off;  // stats + gram accumulators must be zeroed

  float *darr[3], *rne[3], *rna[3];
  for (int l = 0; l < 3; ++l) {
    darr[l] = (float*)(ws + off); off += (size_t)Ns[l] * 4;
    rne[l]  = (float*)(ws + off); off += (size_t)Ns[l] * 4;
    rna[l]  = (float*)(ws + off); off += (size_t)Ns[l] * 4;
  }

  {
    const size_t zn = zero_bytes / 4;
    int zb = (int)((zn + 255) / 256);
    if (zb > 4096) zb = 4096;
    k_zero<<<zb, 256, 0, stream>>>((float*)ws, zn);
  }

  for (int l = 0; l < 3; ++l) {
    const float* fe = (const float*)d_in[2 * l];
    const float* fa = (const float*)d_in[2 * l + 1];
    LvlStats* st = &stats[l];

    k_norm_d<<<Ns[l] / 256, 256, 0, stream>>>(fe, fa, rne[l], rna[l], darr[l],
                                              st, Cs[l], hwShifts[l]);
    k_margin<<<1, 1, 0, stream>>>(st, Ns[l]);
    k_mask<<<Ns[l] / 256, 256, 0, stream>>>(darr[l], rne[l], rna[l], st);

    const int T = Cs[l] / 64;
    const int S = Ns[l] / 1024;  // 32 / 8 / 2  -> 512 workgroups per level
    dim3 grid(T, T, S);
    k_gram<<<grid, 512, 0, stream>>>(fe, fa, rne[l], rna[l],
                                     gram[l][0], gram[l][1], gram[l][2],
                                     Cs[l], hwShifts[l], 1024);

    const int total = Cs[l] * Cs[l];
    int rb = (total + 255) / 256;
    if (rb > 1024) rb = 1024;
    k_gram_reduce<<<rb, 256, 0, stream>>>(gram[l][0], gram[l][1], gram[l][2],
                                          st, total);
  }

  k_final<<<1, 1, 0, stream>>>(&stats[0], &stats[1], &stats[2], (float*)d_out);
}